// ModelfreeGCN_89008902243170
// MI455X (gfx1250) — compile-verified
//
#include <hip/hip_runtime.h>
#include <hip/hip_bf16.h>

// GCNConv (identity weight) = gcn_norm + scatter-add.
// x (51.2MB) and out (51.2MB) both fit in MI455X's 192MB L2, so the random
// gather/scatter traffic is L2-resident. Bottleneck = L2 atomic transaction
// throughput, not HBM (23.3 TB/s sees only ~180MB of one-shot streams).
// CDNA5 paths: saddr-form global loads/atomics via v_readlane-scalarized
// indices, global_atomic_add_f32 arranged so each atomic instruction covers
// ONE contiguous 128B cacheline (1 L2 transaction/instr, 4 per edge),
// non-temporal loads for the streamed edge arrays, global_prefetch_b8
// (__builtin_prefetch) covering all 4 cachelines of next-chunk feature rows.
// WMMA is not applicable: there is no dense matmul in this op (lin == identity).

#define DIM  128
#define DIM4 (DIM / 4)

// ---- kernel 1: deg = 1.0 (self-loop fill value) --------------------------
__global__ __launch_bounds__(256) void k_deg_init(float* __restrict__ deg, int n) {
    int i = blockIdx.x * blockDim.x + threadIdx.x;
    if (i < n) deg[i] = 1.0f;
}

// ---- kernel 2: deg[col[e]] += ew[e] (HW f32 atomics, deg is L2-resident) -
__global__ __launch_bounds__(256) void k_deg_acc(const int* __restrict__ col,
                                                 const float* __restrict__ ew,
                                                 float* __restrict__ deg, int E) {
    int e = blockIdx.x * blockDim.x + threadIdx.x;
    if (e < E) {
        int   c = __builtin_nontemporal_load(&col[e]);   // streamed once: NT
        float w = __builtin_nontemporal_load(&ew[e]);
        unsafeAtomicAdd(&deg[c], w);                     // global_atomic_add_f32
    }
}

// ---- kernel 3: dinv = deg > 0 ? rsqrt(deg) : 0 (needed by edge kernel) ---
__global__ __launch_bounds__(256) void k_dinv(const float* __restrict__ deg,
                                              float* __restrict__ dinv, int n) {
    int i = blockIdx.x * blockDim.x + threadIdx.x;
    if (i < n) {
        float d = deg[i];
        dinv[i] = (d > 0.0f) ? rsqrtf(d) : 0.0f;
    }
}

// ---- kernel 4: out = x[i] / deg[i]  (self-loop term: dinv^2 == 1/deg) ----
// Full overwrite of d_out every call (harness poisons it).
__global__ __launch_bounds__(256) void k_selfloop(const float4* __restrict__ x4,
                                                  const float* __restrict__ deg,
                                                  float4* __restrict__ out4,
                                                  int total /* = n * DIM4 */) {
    int t = blockIdx.x * blockDim.x + threadIdx.x;
    if (t < total) {
        int   i = t >> 5;              // DIM4 == 32 float4 per row
        float d = deg[i];
        float s = (d > 0.0f) ? (1.0f / d) : 0.0f;   // dinv[i] * 1.0 * dinv[i]
        float4 v = x4[t];
        out4[t] = make_float4(s * v.x, s * v.y, s * v.z, s * v.w);
    }
}

// ---- kernel 5: 32-edges-per-wave batched scatter-add ---------------------
// Lane l loads metadata for edge base+l (coalesced NT loads; 2 per-lane dinv
// gathers per 32 edges). Inner loop broadcasts (r, c, norm) for edge k via
// v_readlane -> SGPRs, so gathers/atomics use saddr addressing. Lane l owns
// floats {l, l+32, l+64, l+96} of the 128-wide row, so each of the 4 atomic
// (and load) instructions covers one contiguous 128B line -> 1 L2 transaction
// per instruction. Next chunk's metadata is staged and its 32 feature rows
// fully prefetched (4 x global_prefetch_b8 = 4 x 128B lines) one chunk ahead.
__global__ __launch_bounds__(256) void k_edges(const int* __restrict__ row,
                                               const int* __restrict__ col,
                                               const float* __restrict__ ew,
                                               const float* __restrict__ dinv,
                                               const float* __restrict__ x,
                                               float* __restrict__ out, int E) {
    const int lane   = threadIdx.x & 31;
    const int wave   = (int)((blockIdx.x * blockDim.x + threadIdx.x) >> 5);
    const int nwaves = (int)((gridDim.x * blockDim.x) >> 5);
    const int stride = nwaves * 32;

    int base = wave * 32;
    if (base >= E) return;

    // ---- prologue: chunk 0 metadata + norms + full-row prefetch ----------
    int r = 0, c = 0;
    float nrm = 0.0f;
    {
        const int e = base + lane;
        if (e < E) {
            r = __builtin_nontemporal_load(&row[e]);
            c = __builtin_nontemporal_load(&col[e]);
            const float w = __builtin_nontemporal_load(&ew[e]);
            nrm = dinv[r] * w * dinv[c];
            const char* p = (const char*)(x + (size_t)r * DIM);
            __builtin_prefetch(p +   0, 0, 3);      // global_prefetch_b8
            __builtin_prefetch(p + 128, 0, 3);
            __builtin_prefetch(p + 256, 0, 3);
            __builtin_prefetch(p + 384, 0, 3);
        }
    }

    for (; base < E; base += stride) {
        // ---- stage chunk i+1: metadata loads + prefetches issued early ---
        int rn = 0, cn = 0;
        float nn = 0.0f;
        {
            const int e = base + stride + lane;
            if (e < E) {
                rn = __builtin_nontemporal_load(&row[e]);
                cn = __builtin_nontemporal_load(&col[e]);
                const float w = __builtin_nontemporal_load(&ew[e]);
                nn = dinv[rn] * w * dinv[cn];
                const char* p = (const char*)(x + (size_t)rn * DIM);
                __builtin_prefetch(p +   0, 0, 3);
                __builtin_prefetch(p + 128, 0, 3);
                __builtin_prefetch(p + 256, 0, 3);
                __builtin_prefetch(p + 384, 0, 3);
            }
        }

        // ---- process chunk i: up to 32 edges, broadcast via v_readlane ---
        const int nval = min(32, E - base);
        #pragma unroll 4
        for (int k = 0; k < nval; ++k) {
            const int   rk = __builtin_amdgcn_readlane(r, k);          // SGPR
            const int   ck = __builtin_amdgcn_readlane(c, k);          // SGPR
            const float nk = __int_as_float(
                __builtin_amdgcn_readlane(__float_as_int(nrm), k));    // SGPR

            // cacheline-contiguous ownership: lane l -> {l, l+32, l+64, l+96}
            const float* xr = x + (size_t)rk * DIM + lane;             // saddr + lane*4
            const float v0 = xr[0];                                    // line 0
            const float v1 = xr[32];                                   // line 1
            const float v2 = xr[64];                                   // line 2
            const float v3 = xr[96];                                   // line 3

            float* o = out + (size_t)ck * DIM + lane;                  // saddr + lane*4
            unsafeAtomicAdd(o +  0, nk * v0);   // global_atomic_add_f32, 1 line
            unsafeAtomicAdd(o + 32, nk * v1);   //   (ioffset 128)
            unsafeAtomicAdd(o + 64, nk * v2);   //   (ioffset 256)
            unsafeAtomicAdd(o + 96, nk * v3);   //   (ioffset 384)
        }

        r = rn; c = cn; nrm = nn;
    }
}

extern "C" void kernel_launch(void* const* d_in, const int* in_sizes, int n_in,
                              void* d_out, int out_size, void* d_ws, size_t ws_size,
                              hipStream_t stream) {
    // setup_inputs() order: x [N,128] f32, edge_weight [E] f32, edge_index [2,E] int
    const float* x  = (const float*)d_in[0];
    const float* ew = (const float*)d_in[1];
    const int*   ei = (const int*)d_in[2];     // harness contract: integer -> const int*

    const int E = in_sizes[1];                 // 3,200,000
    const int n = in_sizes[0] / DIM;           // 100,000

    const int* row = ei;                       // edge_index[0]
    const int* col = ei + E;                   // edge_index[1]

    float* out  = (float*)d_out;
    float* deg  = (float*)d_ws;                // n floats
    float* dinv = deg + n;                     // n floats   (800 KB total in d_ws)

    const int B = 256;

    k_deg_init<<<(n + B - 1) / B, B, 0, stream>>>(deg, n);
    k_deg_acc <<<(E + B - 1) / B, B, 0, stream>>>(col, ew, deg, E);
    k_dinv    <<<(n + B - 1) / B, B, 0, stream>>>(deg, dinv, n);

    const int total = n * DIM4;                // 3.2M float4 elements of out
    k_selfloop<<<(total + B - 1) / B, B, 0, stream>>>((const float4*)x, deg,
                                                      (float4*)out, total);

    // 4096 blocks * 8 waves = 32768 waves; each wave handles ~3 chunks of 32
    // edges, so the one-chunk-ahead prefetch pipeline has depth, while 32K
    // independent request streams keep the L2 atomic units saturated.
    k_edges<<<4096, B, 0, stream>>>(row, col, ew, dinv, x, out, E);
}